// RealLRUBPTT_13228499271922
// MI455X (gfx1250) — compile-verified
//
#include <hip/hip_runtime.h>

// RealLRUBPTT forward for MI455X (gfx1250, wave32, WMMA).
//
//   SEQ=2048, BATCH=32, IN_F=256, STATE=512, OUT_F=256
//   b_seq = (x @ B^T) * gamma          GEMM1: M=65536 K=256 N=512  (f32 WMMA)
//   states = scan(lamda, b_seq)        diagonal recurrence, 16384 chains
//   out    = states @ C^T              GEMM2: M=65536 K=512 N=256  (f32 WMMA)
//
// d_out = [output 2048*32*256][final_state 32*512], fp32.
// d_ws  = [lamda 512][gamma 512][b_seq/states 65536*512] floats.

typedef __attribute__((ext_vector_type(2))) float v2f;
typedef __attribute__((ext_vector_type(8))) float v8f;
typedef __attribute__((__vector_size__(4 * sizeof(int)))) int v4i;

#define SEQ   2048
#define BATCH 32
#define IN_F  256
#define STATE 512
#define OUT_F 256
#define MROWS (SEQ * BATCH)   // 65536

// ---- CDNA5 async global->LDS copy path (guarded; sync fallback otherwise) --
#if defined(__AMDGCN__) && __has_builtin(__builtin_amdgcn_global_load_async_to_lds_b128)
#define HAS_ASYNC_LDS 1
#else
#define HAS_ASYNC_LDS 0
#endif

__device__ __forceinline__ void async_copy_b128(const float* g, float* l) {
#if HAS_ASYNC_LDS
  // Param type per hipcc diagnostic: pointer to 16-byte int vector.
  __builtin_amdgcn_global_load_async_to_lds_b128(
      (v4i*)g, (v4i*)l, /*offset=*/0, /*cpol=*/0);
#else
  *(float4*)l = *(const float4*)g;   // global->VGPR->LDS fallback
#endif
}

__device__ __forceinline__ void wait_async_zero() {
#if HAS_ASYNC_LDS
#if __has_builtin(__builtin_amdgcn_s_wait_asynccnt)
  __builtin_amdgcn_s_wait_asynccnt(0);
#else
  asm volatile("s_wait_asynccnt 0x0" ::: "memory");
#endif
#endif
}

// ---------------------------------------------------------------------------
// Kernel 0: lamda = exp(-exp(lambda_log)); gamma = sqrt(1 - lamda^2 + 1e-7)
// ---------------------------------------------------------------------------
__global__ void lru_decay_kernel(const float* __restrict__ lambda_log,
                                 float* __restrict__ lam,
                                 float* __restrict__ gam) {
  const int i = threadIdx.x;           // one block of STATE threads
  const float l = expf(-expf(lambda_log[i]));
  lam[i] = l;
  gam[i] = sqrtf(1.0f - l * l + 1e-7f);
}

// ---------------------------------------------------------------------------
// Shared f32 WMMA GEMM:  Cout[M][N] = A[M][K] @ W[N][K]^T  (* colscale[n])
// Block tile 128(M) x 128(N), K-chunk 32, DOUBLE-BUFFERED via async
// global->LDS copies (ASYNCcnt).  256 threads = 8 waves as 2(M)x4(N);
// each wave owns 4x2 tiles of 16x16 (64 accumulator VGPRs).
//
// ISA frag layout (V_WMMA_F32_16X16X4_F32, wave32):
//   A (16x4):  lane L holds A[m=L&15, k=2*(L>>4)+j] in component j  -> float2
//   B (4x16):  lane L holds B[k=2*(L>>4)+j, n=L&15] in component j  -> float2
//   C (16x16): lane L, vgpr v holds C[m=v+8*(L>>4), n=L&15]
// ---------------------------------------------------------------------------
#define LDS_STR 36   // floats/row: 16B-aligned rows, conflict-free b64 reads

__global__ __launch_bounds__(256)
void lru_gemm_wmma(const float* __restrict__ A,     // [M][K] row-major
                   const float* __restrict__ W,     // [N][K] row-major (B^T cols)
                   float* __restrict__ Cout,        // [M][N] row-major
                   int M, int N, int K,
                   const float* __restrict__ colscale) {  // [N] or nullptr
  __shared__ float As[2][128 * LDS_STR];
  __shared__ float Ws[2][128 * LDS_STR];

  const int tid   = threadIdx.x;
  const int lane  = tid & 31;
  const int wid   = tid >> 5;
  const int wm    = wid >> 2;            // 0..1  -> 64-row slab
  const int wn    = wid & 3;             // 0..3  -> 32-col slab
  const int mlane = lane & 15;
  const int kpair = (lane >> 4) << 1;    // 0 or 2

  const int mbase = blockIdx.y * 128;
  const int nbase = blockIdx.x * 128;

  const v8f vzero = {0.f, 0.f, 0.f, 0.f, 0.f, 0.f, 0.f, 0.f};
  v8f acc[4][2];
#pragma unroll
  for (int i = 0; i < 4; ++i)
#pragma unroll
    for (int j = 0; j < 2; ++j) acc[i][j] = vzero;

  // issue the 128x32 A/W tiles of K-chunk kc into LDS buffer `buf`
  auto issue_chunk = [&](int kc, int buf) {
#pragma unroll
    for (int t = 0; t < 4; ++t) {
      const int idx = tid + 256 * t;     // float4 slot 0..1023
      const int row = idx >> 3;          // 0..127
      const int kq  = (idx & 7) << 2;    // 0,4,...,28
      async_copy_b128(&A[(size_t)(mbase + row) * K + kc + kq],
                      &As[buf][row * LDS_STR + kq]);
      async_copy_b128(&W[(size_t)(nbase + row) * K + kc + kq],
                      &Ws[buf][row * LDS_STR + kq]);
    }
  };

  issue_chunk(0, 0);
  wait_async_zero();
  __syncthreads();

  int cur = 0;
  for (int kc = 0; kc < K; kc += 32) {
    if (kc + 32 < K) issue_chunk(kc + 32, cur ^ 1);  // prefetch next chunk

    const float* as = &As[cur][0];
    const float* ws = &Ws[cur][0];
#pragma unroll
    for (int k0 = 0; k0 < 32; k0 += 4) {
      v2f afrag[4], wfrag[2];
#pragma unroll
      for (int i = 0; i < 4; ++i)
        afrag[i] = *(const v2f*)&as[(wm * 64 + i * 16 + mlane) * LDS_STR + k0 + kpair];
#pragma unroll
      for (int j = 0; j < 2; ++j)
        wfrag[j] = *(const v2f*)&ws[(wn * 32 + j * 16 + mlane) * LDS_STR + k0 + kpair];
#pragma unroll
      for (int i = 0; i < 4; ++i)
#pragma unroll
        for (int j = 0; j < 2; ++j)
          acc[i][j] = __builtin_amdgcn_wmma_f32_16x16x4_f32(
              /*neg_a=*/false, afrag[i], /*neg_b=*/false, wfrag[j],
              /*c_mod=*/(short)0, acc[i][j], /*reuse_a=*/false, /*reuse_b=*/false);
    }

    wait_async_zero();   // own wave's copies into buf^1 done
    __syncthreads();     // everyone's done (dscnt drained by syncthreads)
    cur ^= 1;
  }

  // ---- epilogue: optional per-column scale, scalar stores per C layout
#pragma unroll
  for (int j = 0; j < 2; ++j) {
    const int ncol = nbase + wn * 32 + j * 16 + mlane;
    const float scale = colscale ? colscale[ncol] : 1.0f;
#pragma unroll
    for (int i = 0; i < 4; ++i) {
      const int rbase = mbase + wm * 64 + i * 16 + ((lane >> 4) << 3);
#pragma unroll
      for (int v = 0; v < 8; ++v)
        Cout[(size_t)(rbase + v) * N + ncol] = acc[i][j][v] * scale;
    }
  }
}

// ---------------------------------------------------------------------------
// Kernel 2: in-place diagonal scan   states[s] = lamda*states[s-1] + b_seq[s]
// 4096 threads, one float4 (4 states) each; loads for s+1 are independent of
// the recurrence FMA chain, so they pipeline under unrolling.
// ---------------------------------------------------------------------------
__global__ __launch_bounds__(256)
void lru_scan_kernel(float* __restrict__ bs,            // [SEQ][BATCH][STATE], in-place
                     const float* __restrict__ lam,     // [STATE]
                     float* __restrict__ final_out) {   // [BATCH][STATE]
  const int gid = blockIdx.x * blockDim.x + threadIdx.x;  // 0..4095
  const int b   = gid >> 7;        // batch
  const int nq  = gid & 127;       // state/4 index
  const float4 l = *(const float4*)&lam[nq * 4];
  float4 st = make_float4(0.f, 0.f, 0.f, 0.f);
  float* p = bs + (size_t)b * STATE + (size_t)nq * 4;
  const size_t stride = (size_t)BATCH * STATE;
#pragma unroll 4
  for (int s = 0; s < SEQ; ++s) {
    const float4 v = *(const float4*)p;
    st.x = fmaf(l.x, st.x, v.x);
    st.y = fmaf(l.y, st.y, v.y);
    st.z = fmaf(l.z, st.z, v.z);
    st.w = fmaf(l.w, st.w, v.w);
    *(float4*)p = st;
    p += stride;
  }
  *(float4*)&final_out[(size_t)b * STATE + nq * 4] = st;
}

// ---------------------------------------------------------------------------
extern "C" void kernel_launch(void* const* d_in, const int* in_sizes, int n_in,
                              void* d_out, int out_size, void* d_ws, size_t ws_size,
                              hipStream_t stream) {
  const float* x          = (const float*)d_in[0];  // [SEQ][BATCH][IN_F]
  const float* lambda_log = (const float*)d_in[1];  // [STATE]
  const float* Bw         = (const float*)d_in[2];  // [STATE][IN_F]
  const float* Cw         = (const float*)d_in[3];  // [OUT_F][STATE]
  float* out = (float*)d_out;                        // [SEQ*BATCH*OUT_F | BATCH*STATE]
  float* ws  = (float*)d_ws;

  float* lam = ws;                  // [STATE]
  float* gam = ws + STATE;          // [STATE]
  float* bs  = ws + 2 * STATE;      // [MROWS][STATE]  b_seq -> states (in-place)

  // decay parameters
  lru_decay_kernel<<<1, STATE, 0, stream>>>(lambda_log, lam, gam);

  // GEMM1: b_seq = (x @ B^T) * gamma
  {
    dim3 grid(STATE / 128, MROWS / 128);   // 4 x 512
    lru_gemm_wmma<<<grid, 256, 0, stream>>>(x, Bw, bs, MROWS, STATE, IN_F, gam);
  }

  // scan (in-place), final_state -> tail of d_out
  lru_scan_kernel<<<(BATCH * STATE / 4) / 256, 256, 0, stream>>>(
      bs, lam, out + (size_t)MROWS * OUT_F);

  // GEMM2: out = states @ C^T
  {
    dim3 grid(OUT_F / 128, MROWS / 128);   // 2 x 512
    lru_gemm_wmma<<<grid, 256, 0, stream>>>(bs, Cw, out, MROWS, OUT_F, STATE, nullptr);
  }
}